// DeepBilateralNetCurves_57964878627041
// MI455X (gfx1250) — compile-verified
//
#include <hip/hip_runtime.h>

// ---------------------------------------------------------------------------
// HDRNet forward for MI455X (gfx1250, wave32).
// Phase 1 (lowres, tiny): direct 3x3 convs + f32 WMMA GEMMs for FCs and the
// fused ReLU(global+local) -> 1x1 prediction conv. FC activations are
// pre-packed into a zero-padded [16,K] buffer so the WMMA K-loop has no
// divergent loads (pure global_load_b64 + v_wmma).
// Phase 2 (fullres, memory-bound ~48MB @ 23.3TB/s): single fused kernel,
// guide curve + trilinear slice (sub-grid staged in LDS) + affine apply.
// ---------------------------------------------------------------------------

typedef float v2f __attribute__((ext_vector_type(2)));
typedef float v8f __attribute__((ext_vector_type(8)));

__device__ __forceinline__ int imin(int a, int b) { return a < b ? a : b; }
__device__ __forceinline__ int imax(int a, int b) { return a > b ? a : b; }

// ------------------------- generic direct 3x3 conv -------------------------
__global__ __launch_bounds__(256)
void conv3x3_kernel(const float* __restrict__ in, const float* __restrict__ w,
                    const float* __restrict__ bias, float* __restrict__ out,
                    int B, int Cin, int Hin, int Win,
                    int Cout, int Hout, int Wout, int stride, int relu)
{
    int idx = blockIdx.x * blockDim.x + threadIdx.x;
    int total = B * Cout * Hout * Wout;
    if (idx >= total) return;
    int x  = idx % Wout;
    int y  = (idx / Wout) % Hout;
    int co = (idx / (Wout * Hout)) % Cout;
    int b  = idx / (Wout * Hout * Cout);

    float acc = bias ? bias[co] : 0.0f;
    int yi0 = y * stride - 1, xi0 = x * stride - 1;
    for (int ci = 0; ci < Cin; ++ci) {
        const float* ip = in + ((b * Cin + ci) * Hin) * Win;
        const float* wp = w + (co * Cin + ci) * 9;
#pragma unroll
        for (int ky = 0; ky < 3; ++ky) {
            int yy = yi0 + ky;
            if ((unsigned)yy >= (unsigned)Hin) continue;
#pragma unroll
            for (int kx = 0; kx < 3; ++kx) {
                int xx = xi0 + kx;
                if ((unsigned)xx >= (unsigned)Win) continue;
                acc = fmaf(ip[yy * Win + xx], wp[ky * 3 + kx], acc);
            }
        }
    }
    if (relu) acc = fmaxf(acc, 0.0f);
    out[idx] = acc;
}

// --------------- pack activations into zero-padded [16,K] ------------------
__global__ __launch_bounds__(256)
void pack_a_kernel(const float* __restrict__ A, float* __restrict__ Apad,
                   int Mv, int K)
{
    int idx = blockIdx.x * blockDim.x + threadIdx.x;
    if (idx >= 16 * K) return;
    int m = idx / K, k = idx % K;
    Apad[idx] = (m < Mv) ? A[m * K + k] : 0.0f;
}

// -------------------- FC layer via V_WMMA_F32_16X16X4_F32 -------------------
// out[m,n] = sum_k Apad[m,k] * W[n,k] + bias[n]  (Apad: [16,K], W: [N,K])
// One wave per 16-wide N tile. f32 WMMA layout: A 16x4 (lanes=M, v0:K0|K2,
// v1:K1|K3), B 4x16 mirrored (lanes=N), C/D: vgpr r <-> row r / r+8.
__global__ void fc_wmma_kernel(const float* __restrict__ Apad,
                               const float* __restrict__ W,
                               const float* __restrict__ bias,
                               float* __restrict__ out,
                               int Mv, int K, int N, int relu)
{
    int lane = threadIdx.x & 31;
    int wave = threadIdx.x >> 5;
    int n0 = (blockIdx.x * (blockDim.x >> 5) + wave) * 16;
    if (n0 >= N) return;                       // wave-uniform
    int half = lane >> 4;                      // 0: K+0/K+1, 1: K+2/K+3
    int l = lane & 15;

    const float* arow = Apad + l * K + half * 2;       // 8B-aligned
    const float* brow = W + (n0 + l) * K + half * 2;   // 8B-aligned

    v8f c = {};
    for (int k = 0; k < K; k += 4) {
        v2f a = *(const v2f*)(arow + k);
        v2f b = *(const v2f*)(brow + k);
        c = __builtin_amdgcn_wmma_f32_16x16x4_f32(false, a, false, b,
                                                  (short)0, c, false, false);
    }
#pragma unroll
    for (int r = 0; r < 8; ++r) {
        int m = r + half * 8;
        if (m < Mv) {
            float v = c[r] + (bias ? bias[n0 + l] : 0.0f);
            if (relu) v = fmaxf(v, 0.0f);
            out[m * N + n0 + l] = v;
        }
    }
}

// ------- fusion (ReLU(global+local)) + 1x1 prediction conv via WMMA --------
// Per batch: co[96,256] = pw[96,64] x F[64,256] + pb,  F[k,n]=relu(g[k]+loc[k,n])
__global__ __launch_bounds__(256)
void fuse_pred_wmma_kernel(const float* __restrict__ gfc,   // [B,64]
                           const float* __restrict__ loc,   // [B,64,256]
                           const float* __restrict__ pw,    // [96,64]
                           const float* __restrict__ pb,    // [96]
                           float* __restrict__ co,          // [B,96,256]
                           int B)
{
    int wid  = (blockIdx.x * blockDim.x + threadIdx.x) >> 5;
    int lane = threadIdx.x & 31;
    if (wid >= B * 96) return;                 // wave-uniform
    int nt = wid % 16;
    int mt = (wid / 16) % 6;
    int b  = wid / 96;
    int n0 = nt * 16, m0 = mt * 16;
    int half = lane >> 4, l = lane & 15;

    const float* locb = loc + b * 64 * 256;
    const float* gb   = gfc + b * 64;
    const float* arow = pw + (m0 + l) * 64 + half * 2;
    v8f c = {};
    for (int k = 0; k < 64; k += 4) {
        int ka = k + half * 2;
        v2f a = *(const v2f*)(arow + k);
        v2f bb;
        bb.x = fmaxf(gb[ka]     + locb[ka * 256 + n0 + l], 0.0f);
        bb.y = fmaxf(gb[ka + 1] + locb[(ka + 1) * 256 + n0 + l], 0.0f);
        c = __builtin_amdgcn_wmma_f32_16x16x4_f32(false, a, false, bb,
                                                  (short)0, c, false, false);
    }
#pragma unroll
    for (int r = 0; r < 8; ++r) {
        int m = m0 + r + half * 8;
        co[(b * 96 + m) * 256 + n0 + l] = c[r] + pb[m];
    }
}

// --------------- fused fullres: guide + slice + apply (mem-bound) ----------
// Block = 256 threads handles a 256(x) x 16(y) pixel tile of one batch.
// Grid cells span 64 pixels, so tile needs <= 6 x-cells, <= 3 y-cells,
// all 8 z-levels, 12 channels -> <= 1728 floats staged in LDS as [z][y][x][12].
__global__ __launch_bounds__(256)
void fullres_kernel(const float* __restrict__ img,     // [B,3,H,W]
                    const float* __restrict__ co,      // [B,96,16,16]
                    const float* __restrict__ ccm_w,   // [3,3]
                    const float* __restrict__ ccm_b,   // [3]
                    const float* __restrict__ shifts,  // [3,16]
                    const float* __restrict__ slopes,  // [3,16]
                    const float* __restrict__ proj_w,  // [3]
                    const float* __restrict__ proj_b,  // [1]
                    float* __restrict__ out,           // [B,3,H,W]
                    int H, int W)
{
    __shared__ float sg[8 * 3 * 6 * 12];
    __shared__ float sp[112];  // shifts48 | slopes48 | ccm_w9 | ccm_b3 | pw3 | pb1

    const int b   = blockIdx.z;
    const int x0  = blockIdx.x * 256;
    const int y0  = blockIdx.y * 16;
    const int tid = threadIdx.x;

    if (tid < 48)       sp[tid] = shifts[tid];
    else if (tid < 96)  sp[tid] = slopes[tid - 48];
    else if (tid < 105) sp[tid] = ccm_w[tid - 96];
    else if (tid < 108) sp[tid] = ccm_b[tid - 105];
    else if (tid < 111) sp[tid] = proj_w[tid - 108];
    else if (tid < 112) sp[tid] = proj_b[0];

    const float inv64 = 1.0f / 64.0f;
    int fx_first = (int)floorf((x0 + 0.5f)   * inv64 - 0.5f);
    int fx_last  = (int)floorf((x0 + 255.5f) * inv64 - 0.5f);
    int xi_lo = imax(fx_first, 0);
    int xi_hi = imin(fx_last + 1, 15);
    int nx = xi_hi - xi_lo + 1;
    int fy_first = (int)floorf((y0 + 0.5f)  * inv64 - 0.5f);
    int fy_last  = (int)floorf((y0 + 15.5f) * inv64 - 0.5f);
    int yi_lo = imax(fy_first, 0);
    int yi_hi = imin(fy_last + 1, 15);
    int ny = yi_hi - yi_lo + 1;

    // stage sub-grid: sg[((z*ny+yy)*nx+xx)*12 + c] = grid[b,c,z,y,x] = co[b, z*12+c, y, x]
    const float* cob = co + b * 96 * 256;
    int nelem = 8 * ny * nx * 12;
    for (int i = tid; i < nelem; i += 256) {
        int c  = i % 12;
        int xx = (i / 12) % nx;
        int yy = (i / (12 * nx)) % ny;
        int z  = i / (12 * nx * ny);
        sg[i] = cob[(z * 12 + c) * 256 + (yi_lo + yy) * 16 + (xi_lo + xx)];
    }
    __syncthreads();

    const float* sh  = sp;
    const float* sl  = sp + 48;
    const float* cw  = sp + 96;
    const float* cbv = sp + 105;
    const float* pwj = sp + 108;
    const float  pbj = sp[111];

    int x = x0 + tid;
    float xs  = (x + 0.5f) * inv64 - 0.5f;
    float fxf = floorf(xs);
    int   fx  = (int)fxf;
    float wx1 = xs - fxf, wx0 = 1.0f - wx1;
    int rx0 = imin(imax(fx, 0), 15) - xi_lo;
    int rx1 = imin(imax(fx + 1, 0), 15) - xi_lo;

    const int plane = H * W;
    for (int row = 0; row < 16; ++row) {
        int y = y0 + row;
        float ysv = (y + 0.5f) * inv64 - 0.5f;
        float fyf = floorf(ysv);
        int   fy  = (int)fyf;
        float wy1 = ysv - fyf, wy0 = 1.0f - wy1;
        int ry0 = imin(imax(fy, 0), 15) - yi_lo;
        int ry1 = imin(imax(fy + 1, 0), 15) - yi_lo;

        int pix = (b * 3 * H + y) * W + x;
        float r  = img[pix];
        float g  = img[pix + plane];
        float bl = img[pix + 2 * plane];

        // guide: ccm -> piecewise-linear curve -> projection -> clip
        float gm0 = cw[0] * r + cw[1] * g + cw[2] * bl + cbv[0];
        float gm1 = cw[3] * r + cw[4] * g + cw[5] * bl + cbv[1];
        float gm2 = cw[6] * r + cw[7] * g + cw[8] * bl + cbv[2];
        float v0 = 0.f, v1 = 0.f, v2 = 0.f;
#pragma unroll
        for (int p = 0; p < 16; ++p) {
            v0 = fmaf(sl[p],      fmaxf(gm0 - sh[p],      0.f), v0);
            v1 = fmaf(sl[16 + p], fmaxf(gm1 - sh[16 + p], 0.f), v1);
            v2 = fmaf(sl[32 + p], fmaxf(gm2 - sh[32 + p], 0.f), v2);
        }
        float gm = pwj[0] * v0 + pwj[1] * v1 + pwj[2] * v2 + pbj;
        gm = fminf(fmaxf(gm, 0.f), 1.f);

        float gz  = gm * 8.0f - 0.5f;
        float fzf = floorf(gz);
        int   fz  = (int)fzf;
        float wz1 = gz - fzf, wz0 = 1.0f - wz1;
        int z0 = imin(imax(fz, 0), 7), z1 = imin(imax(fz + 1, 0), 7);

        float acc[12];
#pragma unroll
        for (int c = 0; c < 12; ++c) acc[c] = 0.0f;
        int   zi[2] = { z0, z1 };   float wz[2] = { wz0, wz1 };
        int   yr[2] = { ry0, ry1 }; float wy[2] = { wy0, wy1 };
        int   xr[2] = { rx0, rx1 }; float wx[2] = { wx0, wx1 };
#pragma unroll
        for (int a = 0; a < 2; ++a)
#pragma unroll
            for (int d = 0; d < 2; ++d)
#pragma unroll
                for (int e = 0; e < 2; ++e) {
                    float wgt = wz[a] * wy[d] * wx[e];
                    const float* gp = &sg[((zi[a] * ny + yr[d]) * nx + xr[e]) * 12];
#pragma unroll
                    for (int c = 0; c < 12; ++c) acc[c] = fmaf(wgt, gp[c], acc[c]);
                }

        float o0 = acc[0] * r + acc[1] * g + acc[2]  * bl + acc[3];
        float o1 = acc[4] * r + acc[5] * g + acc[6]  * bl + acc[7];
        float o2 = acc[8] * r + acc[9] * g + acc[10] * bl + acc[11];
        out[pix]             = fminf(fmaxf(o0, 0.f), 1.f);
        out[pix + plane]     = fminf(fmaxf(o1, 0.f), 1.f);
        out[pix + 2 * plane] = fminf(fmaxf(o2, 0.f), 1.f);
    }
}

// ---------------------------------------------------------------------------
extern "C" void kernel_launch(void* const* d_in, const int* in_sizes, int n_in,
                              void* d_out, int out_size, void* d_ws, size_t ws_size,
                              hipStream_t stream)
{
    const float* img_lr = (const float*)d_in[0];
    const float* img_fr = (const float*)d_in[1];
    const float* sw0 = (const float*)d_in[2];  const float* sb0 = (const float*)d_in[3];
    const float* sw1 = (const float*)d_in[4];  const float* sb1 = (const float*)d_in[5];
    const float* sw2 = (const float*)d_in[6];  const float* sb2 = (const float*)d_in[7];
    const float* sw3 = (const float*)d_in[8];  const float* sb3 = (const float*)d_in[9];
    const float* gw0 = (const float*)d_in[10]; const float* gb0 = (const float*)d_in[11];
    const float* gw1 = (const float*)d_in[12]; const float* gb1 = (const float*)d_in[13];
    const float* fw0 = (const float*)d_in[14]; const float* fb0 = (const float*)d_in[15];
    const float* fw1 = (const float*)d_in[16]; const float* fb1 = (const float*)d_in[17];
    const float* fw2 = (const float*)d_in[18]; const float* fb2 = (const float*)d_in[19];
    const float* lw0 = (const float*)d_in[20]; const float* lb0 = (const float*)d_in[21];
    const float* lw1 = (const float*)d_in[22];
    const float* pw  = (const float*)d_in[23]; const float* pb  = (const float*)d_in[24];
    const float* ccm_w  = (const float*)d_in[25]; const float* ccm_b  = (const float*)d_in[26];
    const float* shifts = (const float*)d_in[27]; const float* slopes = (const float*)d_in[28];
    const float* proj_w = (const float*)d_in[29]; const float* proj_b = (const float*)d_in[30];

    const int B = in_sizes[0] / (3 * 256 * 256);
    const int H = 1024, W = 1024;

    float* ws = (float*)d_ws;
    float* t0   = ws;                  size_t o = (size_t)B * 8 * 128 * 128;
    float* t1   = ws + o;              o += (size_t)B * 16 * 64 * 64;
    float* t2   = ws + o;              o += (size_t)B * 32 * 32 * 32;
    float* t3   = ws + o;              o += (size_t)B * 64 * 16 * 16;
    float* g0o  = ws + o;              o += (size_t)B * 64 * 8 * 8;
    float* g1o  = ws + o;              o += (size_t)B * 1024;   // [B,64,4,4] == [B,1024]
    float* fc0o = ws + o;              o += (size_t)B * 256;
    float* fc1o = ws + o;              o += (size_t)B * 128;
    float* fc2o = ws + o;              o += (size_t)B * 64;
    float* l0o  = ws + o;              o += (size_t)B * 64 * 256;
    float* l1o  = ws + o;              o += (size_t)B * 64 * 256;
    float* co   = ws + o;              o += (size_t)B * 96 * 256;
    float* apad = ws + o;              o += (size_t)16 * 1024;  // reused per FC

    auto blocks = [](int n) { return (n + 255) / 256; };

    // splat path 256 -> 128 -> 64 -> 32 -> 16
    int n0 = B * 8 * 128 * 128;
    conv3x3_kernel<<<blocks(n0), 256, 0, stream>>>(img_lr, sw0, sb0, t0, B, 3, 256, 256, 8, 128, 128, 2, 1);
    int n1 = B * 16 * 64 * 64;
    conv3x3_kernel<<<blocks(n1), 256, 0, stream>>>(t0, sw1, sb1, t1, B, 8, 128, 128, 16, 64, 64, 2, 1);
    int n2 = B * 32 * 32 * 32;
    conv3x3_kernel<<<blocks(n2), 256, 0, stream>>>(t1, sw2, sb2, t2, B, 16, 64, 64, 32, 32, 32, 2, 1);
    int n3 = B * 64 * 16 * 16;
    conv3x3_kernel<<<blocks(n3), 256, 0, stream>>>(t2, sw3, sb3, t3, B, 32, 32, 32, 64, 16, 16, 2, 1);

    // global path convs 16 -> 8 -> 4
    int n4 = B * 64 * 8 * 8;
    conv3x3_kernel<<<blocks(n4), 256, 0, stream>>>(t3, gw0, gb0, g0o, B, 64, 16, 16, 64, 8, 8, 2, 1);
    int n5 = B * 64 * 4 * 4;
    conv3x3_kernel<<<blocks(n5), 256, 0, stream>>>(g0o, gw1, gb1, g1o, B, 64, 8, 8, 64, 4, 4, 2, 1);

    // global FCs via f32 WMMA: 1024 -> 256 -> 128 -> 64 (A packed to [16,K])
    pack_a_kernel<<<blocks(16 * 1024), 256, 0, stream>>>(g1o, apad, B, 1024);
    fc_wmma_kernel<<<1, 32 * (256 / 16), 0, stream>>>(apad, fw0, fb0, fc0o, B, 1024, 256, 1);
    pack_a_kernel<<<blocks(16 * 256), 256, 0, stream>>>(fc0o, apad, B, 256);
    fc_wmma_kernel<<<1, 32 * (128 / 16), 0, stream>>>(apad, fw1, fb1, fc1o, B, 256, 128, 1);
    pack_a_kernel<<<blocks(16 * 128), 256, 0, stream>>>(fc1o, apad, B, 128);
    fc_wmma_kernel<<<1, 32 * (64 / 16), 0, stream>>>(apad, fw2, fb2, fc2o, B, 128, 64, 0);

    // local path: two 3x3 convs at 16x16
    conv3x3_kernel<<<blocks(n3), 256, 0, stream>>>(t3,  lw0, lb0,     l0o, B, 64, 16, 16, 64, 16, 16, 1, 1);
    conv3x3_kernel<<<blocks(n3), 256, 0, stream>>>(l0o, lw1, nullptr, l1o, B, 64, 16, 16, 64, 16, 16, 1, 0);

    // fusion + 1x1 prediction via WMMA -> co [B,96,16,16]
    int waves = B * 96;
    fuse_pred_wmma_kernel<<<(waves * 32 + 255) / 256, 256, 0, stream>>>(fc2o, l1o, pw, pb, co, B);

    // fused fullres guide + slice + apply (memory-bound)
    dim3 grid(W / 256, H / 16, B);
    fullres_kernel<<<grid, 256, 0, stream>>>(img_fr, co, ccm_w, ccm_b, shifts, slopes,
                                             proj_w, proj_b, (float*)d_out, H, W);
}